// PolarBPDecoder_61727269978289
// MI455X (gfx1250) — compile-verified
//
#include <hip/hip_runtime.h>

// Polar BP decoder, CDNA5 (gfx1250).
// One workgroup (512 threads = 16 wave32s) per codeword.
// Butterfly stage transitions with lane distance <32 use single-instruction
// DS_SWIZZLE (group-of-32 XOR mode, immediate mask); only cross-wave stages
// (distance 32..256) go through two 4 KB LDS exchange buffers with barriers.
// R messages for the backward pass and L messages for the forward pass are
// register-resident (same thread owns the same index pair at stage s in both
// passes). TDM (tensor_load_to_lds / tensor_store_from_lds) moves the rows.

#define PN       1024
#define PNS      10
#define PITER    10
#define PFROZEN  64
#define PLLRMAX  19.3f
#define KOUT     (PN - PFROZEN)   // 960 info positions, contiguous 64..1023

typedef unsigned int        u32;
typedef unsigned long long  u64;
typedef __attribute__((ext_vector_type(4))) u32 u32x4;
typedef __attribute__((ext_vector_type(8))) u32 u32x8;

// Low 32 bits of a generic shared-aperture address == LDS byte offset.
__device__ __forceinline__ u32 lds_off_of(const void* p) {
  return (u32)(u64)p;
}

// 1-row (2D-degenerate) TDM descriptor; VADDR2/VADDR3 omitted (NULL).
__device__ __forceinline__ void tdm_row(u32 lds_byte_off, u64 gaddr, u32 nelem,
                                        bool is_store) {
  u32x4 g0;
  g0[0] = 1u;                                              // count=1, user D#
  g0[1] = lds_byte_off;                                    // lds_addr (bytes)
  g0[2] = (u32)gaddr;                                      // global_addr[31:0]
  g0[3] = (u32)((gaddr >> 32) & 0x01FFFFFFu) | (2u << 30); // [56:32] | type=2

  u32x8 g1;
  g1[0] = (2u << 16);                        // wg_mask=0, data_size=4B (code 2)
  g1[1] = (nelem & 0xFFFFu) << 16;           // tensor_dim0 lo16 -> bits 63:48
  g1[2] = (nelem >> 16) | (1u << 16);        // tensor_dim0 hi | tensor_dim1=1
  g1[3] = (nelem & 0xFFFFu) << 16;           // tensor_dim1 hi=0 | tile_dim0
  g1[4] = 1u;                                // tile_dim1=1, tile_dim2=0
  g1[5] = nelem;                             // tensor_dim0_stride lo32
  g1[6] = (nelem & 0xFFFFu) << 16;           // stride0 hi=0 | stride1 lo16
  g1[7] = (nelem >> 16);                     // stride1 hi

  if (is_store) {
    asm volatile("tensor_store_from_lds %0, %1" :: "s"(g0), "s"(g1) : "memory");
  } else {
    asm volatile("tensor_load_to_lds %0, %1"    :: "s"(g0), "s"(g1) : "memory");
  }
}

// Stable boxplus: log((1+e^{x+y})/(e^x+e^y)) on clipped inputs.
__device__ __forceinline__ float boxplus(float x, float y) {
  x = fminf(fmaxf(x, -PLLRMAX), PLLRMAX);
  y = fminf(fmaxf(y, -PLLRMAX), PLLRMAX);
  const float a = x + y;
  const float d = x - y;
  const float m = fmaxf(a, 0.0f) - fmaxf(x, y);
  return m + __logf(1.0f + __expf(-fabsf(a)))
           - __logf(1.0f + __expf(-fabsf(d)));
}

// In-wave XOR butterfly via one DS_SWIZZLE (group-of-32: and=0x1F, or=0,
// xor=M). Consumer with bit M of t clear gets (own v1, partner v1); with bit
// set (partner v2, own v2).
template <int M>
__device__ __forceinline__ void bfly_xchg(float v1, float v2, int t,
                                          float& o1, float& o2) {
  const bool bit = (t & M) != 0;
  const float send = bit ? v1 : v2;
  const float recv = __int_as_float(
      __builtin_amdgcn_ds_swizzle(__float_as_int(send), 0x1F | (M << 10)));
  o1 = bit ? recv : v1;
  o2 = bit ? v2 : recv;
}

// ---------------- forward stage S: R messages ----------------
template <int S>
__device__ __forceinline__ void fwd_stage(int t, float* rex,
                                          float (&sr1)[PNS], float (&sr2)[PNS],
                                          const float (&sl1)[PNS - 1],
                                          const float (&sl2)[PNS - 1],
                                          float& p1, float& p2) {
  const int lo = (t << 1) - (t & ((1 << S) - 1));   // IND_1[S][t]
  const int hi = lo + (1 << S);                     // IND_2[S][t]
  float r1, r2;
  if constexpr (S == 0) {
    // msg_r_in: lo=2t, hi=2t+1 are both frozen iff t < PFROZEN/2
    const float f = (t < (PFROZEN / 2)) ? PLLRMAX : 0.0f;
    r1 = f;
    r2 = f;
  } else if constexpr (S <= 5) {                    // distance 1..16: in-wave
    bfly_xchg<(1 << (S - 1))>(p1, p2, t, r1, r2);
  } else {                                          // distance >=32: LDS
    r1 = rex[lo];
    r2 = rex[hi];
  }
  sr1[S] = r1;                                      // keep for backward S
  sr2[S] = r2;
  if constexpr (S < PNS - 1) {                      // r[10] is dead: skip
    const float l1 = sl1[S];                        // l[S+1], saved backward
    const float l2 = sl2[S];
    p1 = boxplus(r1, l2 + r2);                      // r[S+1] at (lo,hi)
    p2 = boxplus(r1, l1) + r2;
    if constexpr (S >= 5) {                         // reader is cross-wave
      rex[lo] = p1;
      rex[hi] = p2;
      __syncthreads();
    }
  }
}

// ---------------- backward stage S: L messages ----------------
template <int S>
__device__ __forceinline__ void bwd_stage(int t, float* lex, float* rex,
                                          const float (&sr1)[PNS],
                                          const float (&sr2)[PNS],
                                          float (&sl1)[PNS - 1],
                                          float (&sl2)[PNS - 1],
                                          float& q1, float& q2,
                                          float ch_lo, float ch_hi, bool last) {
  const int lo = (t << 1) - (t & ((1 << S) - 1));
  const int hi = lo + (1 << S);
  float l1, l2;
  if constexpr (S == PNS - 1) {                     // l[NS] = llr_ch (regs)
    l1 = ch_lo;
    l2 = ch_hi;
  } else if constexpr (S >= 5) {                    // distance >=32: LDS
    l1 = lex[lo];
    l2 = lex[hi];
  } else {                                          // distance 1..16: in-wave
    bfly_xchg<(1 << S)>(q1, q2, t, l1, l2);
  }
  if constexpr (S <= PNS - 2) {                     // reuse in next forward
    sl1[S] = l1;
    sl2[S] = l2;
  }
  const float r1 = sr1[S];                          // r[S] from registers
  const float r2 = sr2[S];
  if constexpr (S >= 1) {
    q1 = boxplus(l1, l2 + r2);                      // l[S] at (lo,hi)
    q2 = boxplus(r1, l1) + l2;
    if constexpr (S >= 6) {                         // reader is cross-wave
      lex[lo] = q1;
      lex[hi] = q2;
      __syncthreads();
    }
  } else {
    if (last) {                                     // l[0]: output only
      const float o1 = boxplus(l1, l2 + r2);
      const float o2 = boxplus(r1, l1) + l2;
      rex[lo] = -o1;                                // lo=2t, hi=2t+1:
      rex[hi] = -o2;                                // stage -l0 for TDM out
      __syncthreads();
    }
  }
}

__global__ __launch_bounds__(512, 1)
void polar_bp_kernel(const float* __restrict__ llr_in, float* __restrict__ out) {
  __shared__ float rex[PN];   // R exchange (fwd 5..8 -> 6..9); TDM stage in/out
  __shared__ float lex[PN];   // L exchange (bwd 9..6 -> 8..5)

  const int t = threadIdx.x;  // butterfly-pair index, 0..511
  const int b = blockIdx.x;   // codeword index

  // --- TDM: pull this codeword's 4 KB LLR row into LDS (wave 0 issues) ---
  if (t < 32) {
    tdm_row(lds_off_of(rex), (u64)(llr_in + (u64)b * PN), PN, false);
    __builtin_amdgcn_s_wait_tensorcnt(0);
  }
  __syncthreads();

  // Channel LLRs live in registers: stage-9 pairing is exactly (t, t+512).
  const float ch_lo = -rex[t];
  const float ch_hi = -rex[t + 512];

  float sr1[PNS], sr2[PNS];         // r[s] at stage-s pairing
  float sl1[PNS - 1], sl2[PNS - 1]; // l[s+1] at stage-s pairing, s=0..8
#pragma unroll
  for (int s = 0; s < PNS - 1; ++s) { sl1[s] = 0.0f; sl2[s] = 0.0f; }
  __syncthreads();   // all rex (llr) reads done before fwd stage 5 overwrites

  for (int it = 0; it < PITER; ++it) {
    const bool last = (it == PITER - 1);

    float p1 = 0.0f, p2 = 0.0f;
    fwd_stage<0>(t, rex, sr1, sr2, sl1, sl2, p1, p2);
    fwd_stage<1>(t, rex, sr1, sr2, sl1, sl2, p1, p2);
    fwd_stage<2>(t, rex, sr1, sr2, sl1, sl2, p1, p2);
    fwd_stage<3>(t, rex, sr1, sr2, sl1, sl2, p1, p2);
    fwd_stage<4>(t, rex, sr1, sr2, sl1, sl2, p1, p2);
    fwd_stage<5>(t, rex, sr1, sr2, sl1, sl2, p1, p2);
    fwd_stage<6>(t, rex, sr1, sr2, sl1, sl2, p1, p2);
    fwd_stage<7>(t, rex, sr1, sr2, sl1, sl2, p1, p2);
    fwd_stage<8>(t, rex, sr1, sr2, sl1, sl2, p1, p2);
    fwd_stage<9>(t, rex, sr1, sr2, sl1, sl2, p1, p2);

    float q1 = 0.0f, q2 = 0.0f;
    bwd_stage<9>(t, lex, rex, sr1, sr2, sl1, sl2, q1, q2, ch_lo, ch_hi, last);
    bwd_stage<8>(t, lex, rex, sr1, sr2, sl1, sl2, q1, q2, ch_lo, ch_hi, last);
    bwd_stage<7>(t, lex, rex, sr1, sr2, sl1, sl2, q1, q2, ch_lo, ch_hi, last);
    bwd_stage<6>(t, lex, rex, sr1, sr2, sl1, sl2, q1, q2, ch_lo, ch_hi, last);
    bwd_stage<5>(t, lex, rex, sr1, sr2, sl1, sl2, q1, q2, ch_lo, ch_hi, last);
    bwd_stage<4>(t, lex, rex, sr1, sr2, sl1, sl2, q1, q2, ch_lo, ch_hi, last);
    bwd_stage<3>(t, lex, rex, sr1, sr2, sl1, sl2, q1, q2, ch_lo, ch_hi, last);
    bwd_stage<2>(t, lex, rex, sr1, sr2, sl1, sl2, q1, q2, ch_lo, ch_hi, last);
    bwd_stage<1>(t, lex, rex, sr1, sr2, sl1, sl2, q1, q2, ch_lo, ch_hi, last);
    bwd_stage<0>(t, lex, rex, sr1, sr2, sl1, sl2, q1, q2, ch_lo, ch_hi, last);
  }

  // --- TDM: push -l0[64..1023] (960 contiguous floats) to the output row ---
  if (t < 32) {
    tdm_row(lds_off_of(rex) + PFROZEN * 4u, (u64)(out + (u64)b * KOUT), KOUT, true);
    // S_ENDPGM performs an implicit wait-idle, covering TENSORcnt.
  }
}

extern "C" void kernel_launch(void* const* d_in, const int* in_sizes, int n_in,
                              void* d_out, int out_size, void* d_ws, size_t ws_size,
                              hipStream_t stream) {
  (void)n_in; (void)out_size; (void)d_ws; (void)ws_size;
  const float* llr = (const float*)d_in[0];
  float* out = (float*)d_out;
  const int bs = in_sizes[0] / PN;   // batch = 512
  polar_bp_kernel<<<bs, 512, 0, stream>>>(llr, out);
}